// LearnableQueryMultiHeadAttention_35055523070638
// MI455X (gfx1250) — compile-verified
//
#include <hip/hip_runtime.h>
#include <math.h>

typedef __attribute__((ext_vector_type(2))) float v2f;
typedef __attribute__((ext_vector_type(8))) float v8f;

#define B_  16
#define L_  4096
#define D_  1024
#define H_  16
#define DK_ 64
#define DV_ 64

// ---------------------------------------------------------------------------
// 16x16 tile GEMM partial-accumulator using V_WMMA_F32_16X16X4_F32.
//   C[16x16] = A[16xKlen] (row-major, lda) * B[Klenx16] (row-major, ldb)
// Callers pre-offset A/B by the K-chunk start. One wave per tile.
// ISA 7.12.2 layouts:
//   A: lanes 0-15 -> M=lane, VGPR{0,1}=K+{0,1}; lanes 16-31 -> K+{2,3}
//   B: lanes 0-15 -> N=lane, rows K+{0,1};      lanes 16-31 -> rows K+{2,3}
//   C: VGPR r -> row r (lanes 0-15) / row r+8 (lanes 16-31), col = lane&15
// ---------------------------------------------------------------------------
static __device__ __forceinline__ v8f wmma_gemm_f32(const float* __restrict__ A, int lda,
                                                    const float* __restrict__ Bm, int ldb,
                                                    int Klen)
{
    const int lane = threadIdx.x & 31;
    const int half = lane >> 4;     // 0 or 1
    const int mn   = lane & 15;

    const float* ap = A  + (size_t)mn * lda + 2 * half;
    const float* bp = Bm + (size_t)(2 * half) * ldb + mn;

    v8f c = {};
    for (int k = 0; k < Klen; k += 4) {
        if ((k & 511) == 0) {
            // speculative streaming prefetch (global_prefetch_b8); OOB is dropped
            __builtin_prefetch(ap + 512, 0, 1);
        }
        v2f a, b;
        a.x = ap[0];
        a.y = ap[1];
        b.x = bp[0];
        b.y = bp[ldb];
        c = __builtin_amdgcn_wmma_f32_16x16x4_f32(false, a, false, b,
                                                  (short)0, c, false, false);
        ap += 4;
        bp += (size_t)4 * ldb;
    }
    return c;
}

// ---------------------------------------------------------------------------
// Tiny setup kernels (negligible cost)
// ---------------------------------------------------------------------------

// qs[h*64+dk] = (query . w_qs[:, h*64+dk]) / sqrt(DK)
__global__ void k_qproj(const float* __restrict__ query, const float* __restrict__ w_qs,
                        float* __restrict__ qs)
{
    int i = blockIdx.x * blockDim.x + threadIdx.x;          // 0..1023
    float acc = 0.f;
    for (int d = 0; d < D_; ++d) acc += query[d] * w_qs[(size_t)d * (H_ * DK_) + i];
    qs[i] = acc * 0.125f;                                   // 1/sqrt(64)
}

// w_eff[d*H + h] = w_ks[d, h*64:(h+1)*64] . qs[h*64:(h+1)*64]
__global__ void k_weff(const float* __restrict__ w_ks, const float* __restrict__ qs,
                       float* __restrict__ w_eff)
{
    int i = blockIdx.x * blockDim.x + threadIdx.x;          // 0..16383
    int d = i >> 4, h = i & 15;
    const float* wk = w_ks + (size_t)d * (H_ * DK_) + h * DK_;
    const float* qh = qs + h * DK_;
    float acc = 0.f;
#pragma unroll
    for (int j = 0; j < DK_; ++j) acc += wk[j] * qh[j];
    w_eff[i] = acc;
}

// ---------------------------------------------------------------------------
// Pass 1: scores[b,h,l] = x[b,l,:] . w_eff[:,h]   (streams x once, 256 MB)
// One wave per (b, l-tile of 16): [16 x 1024] @ [1024 x 16] ; 4096 waves
// Written into the attn region of d_out (softmaxed in place later).
// ---------------------------------------------------------------------------
__global__ void k_scores(const float* __restrict__ x, const float* __restrict__ w_eff,
                         float* __restrict__ attn)
{
    int wave = (blockIdx.x * blockDim.x + threadIdx.x) >> 5;   // 0..4095
    int b  = wave >> 8;                                        // 256 l-tiles per b
    int lt = wave & 255;

    const float* A = x + ((size_t)b * L_ + (size_t)lt * 16) * D_;
    v8f c = wmma_gemm_f32(A, D_, w_eff, H_, D_);

    const int lane = threadIdx.x & 31, half = lane >> 4, n = lane & 15;
    // C(m = l_local, n = h) -> attn[(b*H + h)*L + lt*16 + m]
    float* out = attn + ((size_t)b * H_ + n) * L_ + (size_t)lt * 16 + 8 * half;
#pragma unroll
    for (int r = 0; r < 8; ++r) out[r] = c[r];
}

// ---------------------------------------------------------------------------
// Softmax over L per (b,h) row, in place. One 256-thread block per row.
// ---------------------------------------------------------------------------
__global__ void k_softmax(float* __restrict__ attn)
{
    __shared__ float red[256];
    float* p = attn + (size_t)blockIdx.x * L_;
    const int t = threadIdx.x;

    float vals[16];
    float m = -3.4e38f;
#pragma unroll
    for (int i = 0; i < 16; ++i) { vals[i] = p[t + i * 256]; m = fmaxf(m, vals[i]); }

    red[t] = m; __syncthreads();
    for (int s = 128; s > 0; s >>= 1) {
        if (t < s) red[t] = fmaxf(red[t], red[t + s]);
        __syncthreads();
    }
    m = red[0]; __syncthreads();

    float sum = 0.f;
#pragma unroll
    for (int i = 0; i < 16; ++i) { vals[i] = expf(vals[i] - m); sum += vals[i]; }

    red[t] = sum; __syncthreads();
    for (int s = 128; s > 0; s >>= 1) {
        if (t < s) red[t] += red[t + s];
        __syncthreads();
    }
    float inv = 1.f / red[0];
#pragma unroll
    for (int i = 0; i < 16; ++i) p[t + i * 256] = vals[i] * inv;
}

// ---------------------------------------------------------------------------
// Pass 2 (K-split x4): ctx_x[b,h,:] = attn[b,h,:] @ x[b]  (streams x, 256 MB)
// Per batch: [16 heads x 4096] @ [4096 x 1024]; one wave per
// (b, d-tile of 16, K-chunk of 1024) -> 4096 waves, partials in ws.
// ---------------------------------------------------------------------------
#define CTXX_SPLIT 4
__global__ void k_ctxx_split(const float* __restrict__ attn, const float* __restrict__ x,
                             float* __restrict__ ctx_x_part)
{
    int wave = (blockIdx.x * blockDim.x + threadIdx.x) >> 5;   // 0..4095
    int b  = wave >> 8;
    int t  = wave & 255;
    int s  = t >> 6;                                           // K-chunk 0..3
    int dt = t & 63;                                           // d-tile 0..63
    const int k0 = s * (L_ / CTXX_SPLIT);

    const float* A  = attn + (size_t)b * H_ * L_ + k0;                 // lda = L_
    const float* Bm = x + ((size_t)b * L_ + k0) * D_ + dt * 16;        // ldb = D_
    v8f c = wmma_gemm_f32(A, L_, Bm, D_, L_ / CTXX_SPLIT);

    const int lane = threadIdx.x & 31, half = lane >> 4, n = lane & 15;
    float* out = ctx_x_part + (size_t)s * (B_ * H_ * D_)
               + (size_t)b * H_ * D_ + (size_t)dt * 16 + n;
#pragma unroll
    for (int r = 0; r < 8; ++r) out[(size_t)(r + 8 * half) * D_] = c[r];
}

// Generic deterministic partial-sum reduction: dst[i] = sum_p part[p*n + i]
__global__ void k_reduce(const float* __restrict__ part, float* __restrict__ dst,
                         int nparts, int n)
{
    int stride = gridDim.x * blockDim.x;
    for (int i = blockIdx.x * blockDim.x + threadIdx.x; i < n; i += stride) {
        float s = 0.f;
        for (int p = 0; p < nparts; ++p) s += part[(size_t)p * n + i];
        dst[i] = s;
    }
}

// ---------------------------------------------------------------------------
// ctx[b, h*64+dv] = ctx_x[b,h,:] . w_vs[:, h*64+dv]
// K-split x8: one wave per (h, dv-tile, K-chunk of 128) -> 512 waves
// ---------------------------------------------------------------------------
#define VSPLIT 8
__global__ void k_ctx_split(const float* __restrict__ ctx_x, const float* __restrict__ w_vs,
                            float* __restrict__ ctx_part)
{
    int wave = (blockIdx.x * blockDim.x + threadIdx.x) >> 5;   // 0..511
    int tile = wave >> 3;                                      // 0..63
    int s    = wave & 7;
    int h    = tile >> 2;
    int dvt  = tile & 3;
    const int k0 = s * (D_ / VSPLIT);

    const float* A  = ctx_x + (size_t)h * D_ + k0;                       // lda = H_*D_
    const float* Bm = w_vs + (size_t)k0 * (H_ * DV_) + h * DV_ + dvt * 16; // ldb = H_*DV_
    v8f c = wmma_gemm_f32(A, H_ * D_, Bm, H_ * DV_, D_ / VSPLIT);

    const int lane = threadIdx.x & 31, half = lane >> 4, n = lane & 15;
    float* out = ctx_part + (size_t)s * (B_ * H_ * DV_)
               + h * DV_ + dvt * 16 + n;                       // row m = b, ldc = 1024
#pragma unroll
    for (int r = 0; r < 8; ++r) out[(size_t)(r + 8 * half) * (H_ * DV_)] = c[r];
}

// ---------------------------------------------------------------------------
// out_pre_part[s][b,:] partial of ctx[b,:] @ fc ; K-split x8 -> 512 waves
// ---------------------------------------------------------------------------
__global__ void k_fc_split(const float* __restrict__ ctx, const float* __restrict__ fc,
                           float* __restrict__ out_pre_part)
{
    int wave = (blockIdx.x * blockDim.x + threadIdx.x) >> 5;   // 0..511
    int dt = wave >> 3;                                        // 0..63
    int s  = wave & 7;
    const int k0 = s * ((H_ * DV_) / VSPLIT);

    const float* A  = ctx + k0;                                // lda = H_*DV_
    const float* Bm = fc + (size_t)k0 * D_ + dt * 16;          // ldb = D_
    v8f c = wmma_gemm_f32(A, H_ * DV_, Bm, D_, (H_ * DV_) / VSPLIT);

    const int lane = threadIdx.x & 31, half = lane >> 4, n = lane & 15;
    float* out = out_pre_part + (size_t)s * (B_ * D_) + dt * 16 + n;
#pragma unroll
    for (int r = 0; r < 8; ++r) out[(size_t)(r + 8 * half) * D_] = c[r];
}

// ---------------------------------------------------------------------------
// Fused: sum fc partials + residual(query) + LayerNorm. One block per b.
// ---------------------------------------------------------------------------
__global__ void k_ln(const float* __restrict__ out_pre_part, const float* __restrict__ query,
                     const float* __restrict__ gamma, const float* __restrict__ beta,
                     float* __restrict__ out)
{
    __shared__ float rs[256], rs2[256];
    const int b = blockIdx.x, t = threadIdx.x;

    float vals[4];
    float s = 0.f, s2 = 0.f;
#pragma unroll
    for (int i = 0; i < 4; ++i) {
        const int idx = t + i * 256;
        float v = query[idx];
        for (int p = 0; p < VSPLIT; ++p)
            v += out_pre_part[(size_t)p * (B_ * D_) + (size_t)b * D_ + idx];
        vals[i] = v; s += v; s2 += v * v;
    }
    rs[t] = s; rs2[t] = s2; __syncthreads();
    for (int k = 128; k > 0; k >>= 1) {
        if (t < k) { rs[t] += rs[t + k]; rs2[t] += rs2[t + k]; }
        __syncthreads();
    }
    float mean = rs[0] * (1.f / D_);
    float var  = rs2[0] * (1.f / D_) - mean * mean;
    float inv  = rsqrtf(var + 1e-6f);
#pragma unroll
    for (int i = 0; i < 4; ++i) {
        int idx = t + i * 256;
        out[(size_t)b * D_ + idx] = (vals[i] - mean) * inv * gamma[idx] + beta[idx];
    }
}

// ---------------------------------------------------------------------------
extern "C" void kernel_launch(void* const* d_in, const int* in_sizes, int n_in,
                              void* d_out, int out_size, void* d_ws, size_t ws_size,
                              hipStream_t stream)
{
    (void)in_sizes; (void)n_in; (void)out_size; (void)ws_size;

    const float* x     = (const float*)d_in[0];
    const float* query = (const float*)d_in[1];
    const float* w_qs  = (const float*)d_in[2];
    const float* w_ks  = (const float*)d_in[3];
    const float* w_vs  = (const float*)d_in[4];
    const float* fc    = (const float*)d_in[5];
    const float* gamma = (const float*)d_in[6];
    const float* beta  = (const float*)d_in[7];

    float* out  = (float*)d_out;                 // [B,1,D] = 16384 floats
    float* attn = out + (size_t)B_ * D_;         // [B,H,1,L] = 1048576 floats

    float* ws           = (float*)d_ws;
    float* qs           = ws;                                        //    1024
    float* w_eff        = qs + H_ * DK_;                             //   16384
    float* ctx_x_part   = w_eff + D_ * H_;                           // 1048576 (4 parts)
    float* ctx_x        = ctx_x_part + (size_t)CTXX_SPLIT * B_ * H_ * D_; // 262144
    float* ctx_part     = ctx_x + (size_t)B_ * H_ * D_;              //  131072 (8 parts)
    float* ctx          = ctx_part + (size_t)VSPLIT * B_ * H_ * DV_; //   16384
    float* out_pre_part = ctx + (size_t)B_ * H_ * DV_;               //  131072 (8 parts)

    // 1) q projection + scale
    k_qproj<<<4, 256, 0, stream>>>(query, w_qs, qs);
    // 2) effective key weights (Wk folded with scaled q)
    k_weff<<<64, 256, 0, stream>>>(w_ks, qs, w_eff);
    // 3) scores = x @ w_eff  (4096 waves, streams x)
    k_scores<<<512, 256, 0, stream>>>(x, w_eff, attn);
    // 4) softmax in place over L per (b,h)
    k_softmax<<<B_ * H_, 256, 0, stream>>>(attn);
    // 5) ctx_x partials = attn @ x  (4096 waves, streams x)
    k_ctxx_split<<<512, 256, 0, stream>>>(attn, x, ctx_x_part);
    //    reduce 4 partials -> ctx_x
    k_reduce<<<256, 256, 0, stream>>>(ctx_x_part, ctx_x, CTXX_SPLIT, B_ * H_ * D_);
    // 6) ctx partials = ctx_x @ Wv (per head)  (512 waves)
    k_ctx_split<<<64, 256, 0, stream>>>(ctx_x, w_vs, ctx_part);
    //    reduce 8 partials -> ctx
    k_reduce<<<16, 256, 0, stream>>>(ctx_part, ctx, VSPLIT, B_ * H_ * DV_);
    // 7) out_pre partials = ctx @ fc  (512 waves)
    k_fc_split<<<64, 256, 0, stream>>>(ctx, fc, out_pre_part);
    // 8) fused partial-sum + residual + layernorm -> out
    k_ln<<<B_, 256, 0, stream>>>(out_pre_part, query, gamma, beta, out);
}